// UnsupervisedLoss_SuperResolusion_73332271612422
// MI455X (gfx1250) — compile-verified
//
#include <hip/hip_runtime.h>
#include <stdint.h>

// ---------------------------------------------------------------------------
// MI455X (gfx1250) implementation.
// Bandwidth-bound kernel (400MB logits @ 23.3 TB/s ~= 18us floor), so the
// CDNA5 path used here is the async global->LDS DMA engine (ASYNCcnt) for
// double-buffered tile staging, not WMMA (no matmul structure exists).
// ---------------------------------------------------------------------------

#define N_PIXELS   (1024 * 1024)
#define N_CLASSES  100
#define N_SEGMENTS 30

// ce kernel geometry
#define CE_GRID        1024
#define CE_BLOCK       128                   // 4 waves of 32 (wave32)
#define WAVES_PER_BLK  4
#define TILE_PIX       16                    // pixels staged per wave-tile
#define TILE_FLOATS    (TILE_PIX * N_CLASSES)      // 1600 floats
#define TILE_BYTES     (TILE_FLOATS * 4)           // 6400 bytes
#define CHUNKS         (TILE_BYTES / 256)          // 25 x b64 issues (32 lanes x 8B)
#define N_TILES        (N_PIXELS / TILE_PIX)       // 65536
#define TPW            (N_TILES / (CE_GRID * WAVES_PER_BLK))  // 16 tiles / wave
#define CE_LDS_BYTES   (WAVES_PER_BLK * 2 * TILE_BYTES)       // 51200 (<64KB default)

// ---------------------------------------------------------------------------
// ws layout: [0,3000) int counts | [3000,3030) int mode | [3030] float nll acc
// ---------------------------------------------------------------------------

__global__ __launch_bounds__(256) void init_ws_kernel(int* __restrict__ counts,
                                                      float* __restrict__ acc) {
    int i = blockIdx.x * blockDim.x + threadIdx.x;
    if (i < N_SEGMENTS * N_CLASSES) counts[i] = 0;
    if (i == 0) *acc = 0.0f;
}

// per-block LDS histogram (ds_add_u32), merged with global atomics
__global__ __launch_bounds__(256) void hist_kernel(const int* __restrict__ target,
                                                   const int* __restrict__ seg,
                                                   int* __restrict__ counts) {
    __shared__ int lh[N_SEGMENTS * N_CLASSES];       // 12 KB
    for (int i = threadIdx.x; i < N_SEGMENTS * N_CLASSES; i += blockDim.x) lh[i] = 0;
    __syncthreads();
    const int stride = gridDim.x * blockDim.x;
    for (int i = blockIdx.x * blockDim.x + threadIdx.x; i < N_PIXELS; i += stride) {
        atomicAdd(&lh[seg[i] * N_CLASSES + target[i]], 1);
    }
    __syncthreads();
    for (int i = threadIdx.x; i < N_SEGMENTS * N_CLASSES; i += blockDim.x) {
        int v = lh[i];
        if (v) atomicAdd(&counts[i], v);
    }
}

// jnp.argmax semantics: first (lowest class id) maximum wins -> strict '>'
__global__ void mode_kernel(const int* __restrict__ counts, int* __restrict__ mode) {
    int s = threadIdx.x;
    if (s < N_SEGMENTS) {
        int best = 0, bi = 0;
        for (int c = 0; c < N_CLASSES; ++c) {
            int v = counts[s * N_CLASSES + c];
            if (v > best) { best = v; bi = c; }
        }
        mode[s] = bi;
    }
}

// Issue one 6400B tile as 25 async b64 copies: each issue moves 32 lanes x 8B.
// VDST operand = per-lane LDS byte address, VADDR = per-lane 64-bit global addr.
__device__ __forceinline__ void issue_tile_async(const float* gbase,
                                                 unsigned ldsBase, int lane) {
    const char* ga = (const char*)gbase + lane * 8;
    unsigned     la = ldsBase + (unsigned)(lane * 8);
#pragma unroll
    for (int j = 0; j < CHUNKS; ++j) {
        asm volatile("global_load_async_to_lds_b64 %0, %1, off"
                     :: "v"(la), "v"(ga)
                     : "memory");
        ga += 256;
        la += 256;
    }
}

__global__ __launch_bounds__(CE_BLOCK) void ce_relabel_kernel(
    const float* __restrict__ logits,
    const int*   __restrict__ target,
    const int*   __restrict__ seg,
    const int*   __restrict__ mode,
    float*       __restrict__ acc,
    float*       __restrict__ dout) {

    extern __shared__ float smem[];                  // 51200B: 4 waves x 2 bufs x 6400B
    const int lane  = threadIdx.x & 31;
    const int wib   = threadIdx.x >> 5;              // wave in block: 0..3
    const int waveG = blockIdx.x * WAVES_PER_BLK + wib;

    // low 32 bits of a generic LDS pointer == workgroup-relative LDS byte offset
    const unsigned ldsBase = (unsigned)(uint64_t)(uintptr_t)smem
                           + (unsigned)(wib * 2 * TILE_BYTES);
    const int smemF = wib * 2 * TILE_FLOATS;

    const int tile0 = waveG * TPW;                   // 16 contiguous tiles per wave

    // prefetch tile 0 into buffer 0 (ASYNCcnt = 25)
    issue_tile_async(logits + (size_t)tile0 * TILE_FLOATS, ldsBase, lane);

    const int p  = lane & 15;                        // pixel within tile
    const int hf = lane >> 4;                        // class-half: 0 -> [0,50), 1 -> [50,100)
    float nllAcc = 0.0f;

    for (int k = 0; k < TPW; ++k) {
        const int buf = k & 1;
        if (k + 1 < TPW) {
            // prefetch next tile into the other buffer, then wait until only
            // those 25 remain outstanding (async loads complete in order).
            issue_tile_async(logits + (size_t)(tile0 + k + 1) * TILE_FLOATS,
                             ldsBase + (unsigned)(((k + 1) & 1) * TILE_BYTES), lane);
            asm volatile("s_wait_asynccnt 25" ::: "memory");
        } else {
            asm volatile("s_wait_asynccnt 0" ::: "memory");
        }

        // Bank-conflict-free: float idx = 100*p + 50*hf + c; the two halves
        // occupy disjoint residues mod 4, and 36p mod 64 is injective on p<16.
        const float* row = &smem[smemF + buf * TILE_FLOATS + p * N_CLASSES + hf * 50];

        float m = -1e30f;
#pragma unroll
        for (int c = 0; c < 50; ++c) m = fmaxf(m, row[c]);
        m = fmaxf(m, __shfl_xor(m, 16, 32));         // combine the two halves

        float s = 0.0f;
#pragma unroll
        for (int c = 0; c < 50; ++c) s += __expf(row[c] - m);
        s += __shfl_xor(s, 16, 32);

        const int pix = (tile0 + k) * TILE_PIX + p;
        if (hf == 0) {
            const int   t  = target[pix];
            const float xt = smem[smemF + buf * TILE_FLOATS + p * N_CLASSES + t];
            nllAcc += m + __logf(s) - xt;            // -log_softmax at target
            dout[1 + pix] = (float)mode[seg[pix]];   // relabeled
        }
    }

    // wave32 reduction, one global atomic per wave
#pragma unroll
    for (int off = 16; off > 0; off >>= 1) nllAcc += __shfl_xor(nllAcc, off, 32);
    if (lane == 0) atomicAdd(acc, nllAcc);
}

__global__ void finalize_kernel(const float* __restrict__ acc,
                                float* __restrict__ dout) {
    if (threadIdx.x == 0 && blockIdx.x == 0) {
        dout[0] = *acc * (1.0f / (float)N_PIXELS);
    }
}

// ---------------------------------------------------------------------------

extern "C" void kernel_launch(void* const* d_in, const int* in_sizes, int n_in,
                              void* d_out, int out_size, void* d_ws, size_t ws_size,
                              hipStream_t stream) {
    const float* logits = (const float*)d_in[0];   // [N_PIXELS, N_CLASSES] f32
    const int*   target = (const int*)d_in[1];     // [N_PIXELS] int32
    const int*   seg    = (const int*)d_in[2];     // [N_PIXELS] int32
    float*       dout   = (float*)d_out;           // [0]=loss, [1..N]=relabeled

    int*   counts = (int*)d_ws;
    int*   mode   = counts + N_SEGMENTS * N_CLASSES;
    float* acc    = (float*)(counts + N_SEGMENTS * N_CLASSES + N_SEGMENTS);

    (void)in_sizes; (void)n_in; (void)out_size; (void)ws_size;

    init_ws_kernel<<<12, 256, 0, stream>>>(counts, acc);
    hist_kernel<<<256, 256, 0, stream>>>(target, seg, counts);
    mode_kernel<<<1, 32, 0, stream>>>(counts, mode);
    ce_relabel_kernel<<<CE_GRID, CE_BLOCK, CE_LDS_BYTES, stream>>>(
        logits, target, seg, mode, acc, dout);
    finalize_kernel<<<1, 1, 0, stream>>>(acc, dout);
}